// CustomLoss_div_70729521430607
// MI455X (gfx1250) — compile-verified
//
#include <hip/hip_runtime.h>
#include <hip/hip_bf16.h>

typedef __attribute__((ext_vector_type(2))) float v2f;
typedef __attribute__((ext_vector_type(8))) float v8f;

#define S_CH 4194304L              // 256*256*64, channel stride
#define Q4   1048576L              // S_CH/4, channel stride in float4
#define N_PT 8388608.0f            // 2*256*256*64 pointwise elements
#define N_DIV 8190450.0f           // 2*255*255*63 interior cells
#define EPS 1e-10f
#define DXc 0.1f
#define DYc 0.1f

// ---------------- Kernel 1: pointwise losses (loss_b, loss_parallel) ----------------
__global__ void cl_pointwise(const float* __restrict__ outputs,
                             const float* __restrict__ targets,
                             float* __restrict__ ws, int G1) {
    const float4* o4 = (const float4*)outputs;
    const float4* t4 = (const float4*)targets;
    const long total4 = 2L * Q4;                // 2,097,152 float4 cells
    float sb1 = 0.f, sb2 = 0.f, sp = 0.f;
    const long stride = (long)gridDim.x * blockDim.x;
    for (long e = (long)blockIdx.x * blockDim.x + threadIdx.x; e < total4; e += stride) {
        const long n = e / Q4, r = e % Q4;
        const float4 bxp = o4[n * 3 * Q4 + r];
        const float4 byp = o4[n * 3 * Q4 + Q4 + r];
        const float4 bzp = o4[n * 3 * Q4 + 2 * Q4 + r];
        const float4 bxt = t4[n * 4 * Q4 + r];
        const float4 byt = t4[n * 4 * Q4 + Q4 + r];
        const float4 bzt = t4[n * 4 * Q4 + 2 * Q4 + r];
#pragma unroll
        for (int c = 0; c < 4; ++c) {
            const float xp = ((const float*)&bxp)[c], yp = ((const float*)&byp)[c];
            const float zp = ((const float*)&bzp)[c];
            const float xt = ((const float*)&bxt)[c], yt = ((const float*)&byt)[c];
            const float zt = ((const float*)&bzt)[c];
            const float t2 = xt * xt + yt * yt;
            const float a = xp * xp + yp * yp - t2;
            sb1 += a * a / (t2 + EPS);
            const float dz = zp - zt;
            const float dz2 = dz * dz;
            sb2 += dz2 * dz2 / (zt * zt + EPS);
            const float cr = xp * yt - yp * xt;
            sp += cr * cr / (t2 + zt * zt + EPS);
        }
    }
    __shared__ float sh[3][256];
    const int tid = threadIdx.x;
    sh[0][tid] = sb1; sh[1][tid] = sb2; sh[2][tid] = sp;
    __syncthreads();
    for (int s = 128; s > 0; s >>= 1) {
        if (tid < s) {
            sh[0][tid] += sh[0][tid + s];
            sh[1][tid] += sh[1][tid + s];
            sh[2][tid] += sh[2][tid + s];
        }
        __syncthreads();
    }
    if (tid == 0) {
        ws[blockIdx.x]          = sh[0][0];
        ws[G1 + blockIdx.x]     = sh[1][0];
        ws[2 * G1 + blockIdx.x] = sh[2][0];
    }
}

// ---------------- Kernel 2: divergence stencil (loss_div numerator sum) ----------------
__global__ void cl_divstencil(const float* __restrict__ outputs,
                              const float* __restrict__ targets,
                              float* __restrict__ ws_div) {
    const long total = 2L * 255L * 255L * 63L;
    float acc = 0.f;
    const long stride = (long)gridDim.x * blockDim.x;
    for (long idx = (long)blockIdx.x * blockDim.x + threadIdx.x; idx < total; idx += stride) {
        long t = idx;
        const int k = (int)(t % 63); t /= 63;
        const int j = (int)(t % 255); t /= 255;
        const int i = (int)(t % 255);
        const int n = (int)(t / 255);
        const long cell = (long)i * 16384 + (long)j * 64 + k;   // strides: i=256*64, j=64, k=1
        const float* bxp = outputs + (long)n * 3 * S_CH + cell;
        const float* byp = bxp + S_CH;
        const float* bzp = bxp + 2 * S_CH;
        const float* zt  = targets + (long)n * 4 * S_CH + 3 * S_CH + cell;
        float bx[2][2][2], by[2][2][2], bz[2][2][2], z[2][2][2];
#pragma unroll
        for (int di = 0; di < 2; ++di)
#pragma unroll
            for (int dj = 0; dj < 2; ++dj)
#pragma unroll
                for (int dk = 0; dk < 2; ++dk) {
                    const long o = (long)di * 16384 + (long)dj * 64 + dk;
                    bx[di][dj][dk] = bxp[o];
                    by[di][dj][dk] = byp[o];
                    bz[di][dj][dk] = bzp[o];
                    z[di][dj][dk]  = zt[o];
                }
        const float sixth = 1.0f / 6.0f;
        float num =
            0.25f*(bx[1][0][0]+bx[1][1][0]+bx[1][0][1]+bx[1][1][1])*DYc*0.5f*(z[1][0][1]-z[1][0][0]+z[1][1][1]-z[1][1][0])
          - 0.25f*(bx[0][0][0]+bx[0][1][0]+bx[0][0][1]+bx[0][1][1])*DYc*0.5f*(z[0][0][1]-z[0][0][0]+z[0][1][1]-z[0][1][0])
          + 0.25f*(by[0][1][0]+by[1][1][0]+by[0][1][1]+by[1][1][1])*DXc*0.5f*(z[0][1][1]-z[0][1][0]+z[1][1][1]-z[1][1][0])
          - 0.25f*(by[0][0][0]+by[1][0][0]+by[0][0][1]+by[1][0][1])*DXc*0.5f*(z[0][0][1]-z[0][0][0]+z[1][0][1]-z[1][0][0])
          + 0.25f*(bz[0][0][1]+bz[0][1][1]+bz[1][0][1]+bz[1][1][1])*DXc*DYc
          - 0.25f*(bz[0][0][0]+bz[0][1][0]+bz[1][0][0]+bz[1][1][0])*DXc*DYc
          + (bx[0][0][1]+bx[1][0][1]+bx[1][1][1])*DYc*(z[0][0][1]-z[1][0][1])*sixth
          + (bx[0][1][1]+bx[1][1][1]+bx[1][0][1])*DYc*(z[0][1][1]-z[1][1][1])*sixth
          + (by[1][0][1]+by[1][1][1]+by[0][1][1])*DXc*(z[1][0][1]-z[1][1][1])*sixth
          + (by[0][0][1]+by[0][1][1]+by[1][1][1])*DXc*(z[0][0][1]-z[0][1][1])*sixth
          - (bx[0][0][0]+bx[1][0][0]+bx[1][1][0])*DYc*(z[0][0][0]-z[1][0][0])*sixth
          - (bx[0][1][0]+bx[1][1][0]+bx[1][0][0])*DYc*(z[0][1][0]-z[1][1][0])*sixth
          - (by[1][0][0]+by[1][1][0]+by[0][1][0])*DXc*(z[1][0][0]-z[1][1][0])*sixth
          - (by[0][0][0]+by[0][1][0]+by[1][1][0])*DXc*(z[0][0][0]-z[0][1][0])*sixth;
        float bxc = 0.f, byc = 0.f, bzc = 0.f;
#pragma unroll
        for (int di = 0; di < 2; ++di)
#pragma unroll
            for (int dj = 0; dj < 2; ++dj)
#pragma unroll
                for (int dk = 0; dk < 2; ++dk) {
                    bxc += bx[di][dj][dk]; byc += by[di][dj][dk]; bzc += bz[di][dj][dk];
                }
        bxc *= 0.125f; byc *= 0.125f; bzc *= 0.125f;
        const float den = bxc * bxc + byc * byc + bzc * bzc + EPS;
        acc += num * num / den;
    }
    __shared__ float sh[256];
    const int tid = threadIdx.x;
    sh[tid] = acc;
    __syncthreads();
    for (int s = 128; s > 0; s >>= 1) {
        if (tid < s) sh[tid] += sh[tid + s];
        __syncthreads();
    }
    if (tid == 0) ws_div[blockIdx.x] = sh[0];
}

// ---------------- Kernel 3: single-wave finalize with WMMA cross-lane reduction ----------------
// Per-lane value v goes into A(16x4) as A[m][0]=v_m (lanes 0-15), A[m][2]=v_{m+16}
// (lanes 16-31); B = ones => D[m][n] = v_m + v_{m+16}. Lane sums its 8 D VGPRs;
// total = t(lane0) + t(lane16).
__device__ __forceinline__ float wave_reduce_wmma(float v) {
    v2f a; a.x = v; a.y = 0.f;
    v2f b; b.x = 1.f; b.y = 1.f;
    v8f c = {};
    c = __builtin_amdgcn_wmma_f32_16x16x4_f32(false, a, false, b, (short)0, c, false, false);
    float t = c[0] + c[1] + c[2] + c[3] + c[4] + c[5] + c[6] + c[7];
    return __shfl(t, 0, 32) + __shfl(t, 16, 32);
}

__global__ void cl_finalize(const float* __restrict__ ws, float* __restrict__ out,
                            int G1, int G2) {
    const int lane = threadIdx.x;          // exactly 32 threads: EXEC all ones for WMMA
    float sb1 = 0.f, sb2 = 0.f, sp = 0.f, sd = 0.f;
    for (int i = lane; i < G1; i += 32) {  // G1 multiple of 32 -> no divergence
        sb1 += ws[i];
        sb2 += ws[G1 + i];
        sp  += ws[2 * G1 + i];
    }
    for (int i = lane; i < G2; i += 32) sd += ws[3 * G1 + i];
    const float tb1 = wave_reduce_wmma(sb1);
    const float tb2 = wave_reduce_wmma(sb2);
    const float tp  = wave_reduce_wmma(sp);
    const float td  = wave_reduce_wmma(sd);
    if (lane == 0) {
        const float loss_b   = tb1 / N_PT + tb2 / N_PT;
        const float loss_par = tp / N_PT;
        const float loss_div = (td / N_DIV) / (DXc * DXc) / (DYc * DYc);
        out[0] = 1000.0f * loss_b + 1000.0f * loss_par;
        out[1] = 100.0f * loss_div;
    }
}

extern "C" void kernel_launch(void* const* d_in, const int* in_sizes, int n_in,
                              void* d_out, int out_size, void* d_ws, size_t ws_size,
                              hipStream_t stream) {
    const float* outputs = (const float*)d_in[0];
    const float* targets = (const float*)d_in[1];
    float* out = (float*)d_out;
    float* ws  = (float*)d_ws;
    const int G1 = 1024, G2 = 1024;
    cl_pointwise<<<G1, 256, 0, stream>>>(outputs, targets, ws, G1);
    cl_divstencil<<<G2, 256, 0, stream>>>(outputs, targets, ws + 3 * G1);
    cl_finalize<<<1, 32, 0, stream>>>(ws, out, G1, G2);
}